// DEQModule_28501402976690
// MI455X (gfx1250) — compile-verified
//
#include <hip/hip_runtime.h>
#include <math.h>

// DEQ Anderson Type-I solver for f(z) = tanh(zW + x + b) on gfx1250.
// GEMM path: V_WMMA_F32_16X16X4_F32 (full-f32 matrix pipe — required for TOL=1e-5).
#define Bsz 2048
#define Dd  1024
#define Mm  5
#define TOLf 1e-5f
#define REGf 1e-6f

typedef float v2f __attribute__((ext_vector_type(2)));
typedef float v8f __attribute__((ext_vector_type(8)));

static __device__ __forceinline__ float blockReduceSum(float v) {
  int lane = threadIdx.x & 31;
  int w    = threadIdx.x >> 5;
  #pragma unroll
  for (int o = 16; o > 0; o >>= 1) v += __shfl_down(v, o, 32);
  __shared__ float s[8];
  __syncthreads();
  if (lane == 0) s[w] = v;
  __syncthreads();
  float r = (threadIdx.x < 8) ? s[threadIdx.x] : 0.0f;
  if (w == 0) {
    #pragma unroll
    for (int o = 4; o > 0; o >>= 1) r += __shfl_down(r, o, 32);
  }
  return r; // valid in thread 0
}

// ---------------------------------------------------------------------------
// Fused GEMM: g_out = tanh(zin*W + x + b) - zin.
// Block tile 64x128, BK=16, double-buffered LDS, one barrier per K-step.
// Wave w: rows [16*(w&3)..+16), cols [64*(w>>2)..+64) -> four 16x16 WMMA accs.
// ---------------------------------------------------------------------------
__global__ __launch_bounds__(256) void gemm_g(
    const int*   __restrict__ gate,   // skip whole kernel if *gate != 0
    const float* __restrict__ zin,
    const float* __restrict__ W,
    const float* __restrict__ x,
    const float* __restrict__ bvec,
    float*       __restrict__ g_out)
{
  if (gate && *gate) return;

  __shared__ __align__(16) float As[2][64][20];   // 64 rows x BK16, padded
  __shared__ __align__(16) float Bs[2][16][132];  // BK16 x 128 cols, padded

  const int tid  = threadIdx.x;
  const int lane = tid & 31;
  const int wv   = tid >> 5;
  const int wrow = wv & 3;     // 16-row group
  const int wcol = wv >> 2;    // 64-col group (0/1)
  const int r16  = lane & 15;
  const int kh   = lane >> 4;  // lane group selects K pair {0,1} vs {2,3}

  const int brow = blockIdx.y; // 0..31
  const int bcol = blockIdx.x; // 0..7

  // A-tile loader: 64x16 floats, one float4 per thread
  const int lr = tid >> 2;          // As row 0..63
  const int lq = (tid & 3) << 2;    // As col quad
  // B-tile loader: 16x128 floats, two float4 per thread
  const int br = tid >> 4;          // Bs row (k) 0..15
  const int bq = (tid & 15) << 3;   // Bs col base (8 cols per thread)

  const float* za = zin + (size_t)(brow * 64 + lr) * Dd + lq;
  const float* wa = W + (size_t)br * Dd + (size_t)bcol * 128 + bq;

  v8f acc[4] = {v8f{}, v8f{}, v8f{}, v8f{}};

  // Preload tile 0
  {
    float4 av  = *(const float4*)(za);
    float4 bv0 = *(const float4*)(wa);
    float4 bv1 = *(const float4*)(wa + 4);
    *(float4*)&As[0][lr][lq]     = av;
    *(float4*)&Bs[0][br][bq]     = bv0;
    *(float4*)&Bs[0][br][bq + 4] = bv1;
  }
  __syncthreads();

  int buf = 0;
  for (int k0 = 0; k0 < Dd; k0 += 16) {
    const int kn = k0 + 16;
    float4 av, bv0, bv1;
    if (kn < Dd) {
      av  = *(const float4*)(za + kn);
      bv0 = *(const float4*)(wa + (size_t)kn * Dd);
      bv1 = *(const float4*)(wa + (size_t)kn * Dd + 4);
      if (kn + 16 < Dd)
        __builtin_prefetch(wa + (size_t)(kn + 16) * Dd, 0, 3);
    }

    // Compute on current buffer
    #pragma unroll
    for (int kk = 0; kk < 16; kk += 4) {
      v2f a;
      a.x = As[buf][wrow * 16 + r16][kk + 2 * kh + 0];
      a.y = As[buf][wrow * 16 + r16][kk + 2 * kh + 1];
      #pragma unroll
      for (int j = 0; j < 4; ++j) {
        const int c = wcol * 64 + 16 * j + r16;
        v2f bf;
        bf.x = Bs[buf][kk + 2 * kh + 0][c];
        bf.y = Bs[buf][kk + 2 * kh + 1][c];
        acc[j] = __builtin_amdgcn_wmma_f32_16x16x4_f32(false, a, false, bf,
                                                       (short)0, acc[j],
                                                       false, false);
      }
    }

    if (kn < Dd) {
      *(float4*)&As[buf ^ 1][lr][lq]     = av;
      *(float4*)&Bs[buf ^ 1][br][bq]     = bv0;
      *(float4*)&Bs[buf ^ 1][br][bq + 4] = bv1;
    }
    __syncthreads();
    buf ^= 1;
  }

  // Epilogue: f = tanh(acc + x + b); g = f - z
  #pragma unroll
  for (int j = 0; j < 4; ++j) {
    const int gcol = bcol * 128 + wcol * 64 + 16 * j + r16;
    #pragma unroll
    for (int r = 0; r < 8; ++r) {
      const int grow = brow * 64 + wrow * 16 + kh * 8 + r;
      size_t i0 = (size_t)grow * Dd + gcol;
      float f0 = tanhf(acc[j][r] + x[i0] + bvec[gcol]);
      g_out[i0] = f0 - zin[i0];
    }
  }
}

// ---------------------------------------------------------------------------
// Row norms: gn[b]=||g[b,:]||, optionally zn[b]=||z[b,:]||. One block per row.
// ---------------------------------------------------------------------------
__global__ __launch_bounds__(256) void rownorm(
    const int* gate, const float* __restrict__ g, const float* __restrict__ z,
    float* __restrict__ gn, float* __restrict__ zn)
{
  if (gate && *gate) return;
  int b = blockIdx.x, t = threadIdx.x;
  const float* gr = g + (size_t)b * Dd;
  float sg = 0.f, sz = 0.f;
  if (z) {
    const float* zr = z + (size_t)b * Dd;
    for (int d = t; d < Dd; d += 256) {
      float a = gr[d]; sg += a * a;
      float c = zr[d]; sz += c * c;
    }
  } else {
    for (int d = t; d < Dd; d += 256) { float a = gr[d]; sg += a * a; }
  }
  sg = blockReduceSum(sg);
  if (z) sz = blockReduceSum(sz);
  if (t == 0) { gn[b] = sqrtf(sg); if (z) zn[b] = sqrtf(sz); }
}

// ---------------------------------------------------------------------------
// Iteration control: res, rel_k, better/bres, halt/done.
// S[0]=bres, S[1]=rel.  SI: 0=done, 1=ptr, 2=halt, 3=better, 4=bad
// ---------------------------------------------------------------------------
__global__ __launch_bounds__(256) void control1(const float* gn, const float* zn,
                                                float* S, int* SI)
{
  float sg = 0.f, sz = 0.f;
  for (int i = threadIdx.x; i < Bsz; i += 256) { sg += gn[i]; sz += zn[i]; }
  sg = blockReduceSum(sg);
  sz = blockReduceSum(sz);
  if (threadIdx.x == 0) {
    float res   = sg / (float)Bsz;
    float meanz = sz / (float)Bsz;
    float relk  = res / (meanz + 1e-9f);
    int done   = SI[0];
    int better = (!done) && (res < S[0]);
    if (better) S[0] = res;
    if (!done)  S[1] = relk;
    int halt = done || (relk < TOLf);
    SI[0] = halt; SI[2] = halt; SI[3] = better;
  }
}

__global__ __launch_bounds__(256) void control2(const int* halt, const float* gn2,
                                                const float* S, int* SI)
{
  if (*halt) return;
  float sg = 0.f;
  for (int i = threadIdx.x; i < Bsz; i += 256) sg += gn2[i];
  sg = blockReduceSum(sg);
  if (threadIdx.x == 0) SI[4] = ((sg / (float)Bsz) > 3.0f * S[0]) ? 1 : 0;
}

__global__ void bz_copy(const int* better, const float* __restrict__ z,
                        float* __restrict__ bz)
{
  if (!*better) return;
  size_t i = (size_t)blockIdx.x * blockDim.x + threadIdx.x;
  if (i < (size_t)Bsz * Dd) bz[i] = z[i];
}

__global__ void colupd(const int* halt, const int* SI,
                       const float* __restrict__ z,  const float* __restrict__ pz,
                       const float* __restrict__ g,  const float* __restrict__ pg,
                       float* __restrict__ dX, float* __restrict__ dG)
{
  if (*halt) return;
  size_t i = (size_t)blockIdx.x * blockDim.x + threadIdx.x;
  if (i >= (size_t)Bsz * Dd) return;
  int col  = SI[1] % Mm;
  size_t b = i / Dd, d = i % Dd;
  size_t o = (b * Mm + col) * Dd + d;
  dX[o] = z[i] - pz[i];
  dG[o] = g[i] - pg[i];
}

// One block per batch row; SKETCH(=256) == blockDim.  Builds GtG/Gtg from the
// sketch columns and solves the regularized 5x5 system.
__global__ __launch_bounds__(256) void sketch_solve(
    const int* halt, const float* __restrict__ dG, const float* __restrict__ g,
    const int* __restrict__ sidx, float* __restrict__ alpha)
{
  if (*halt) return;
  int b = blockIdx.x, t = threadIdx.x;
  int is = sidx[t];
  const float* dGb = dG + (size_t)b * Mm * Dd;
  float gv = g[(size_t)b * Dd + is];
  float dgv[Mm];
  #pragma unroll
  for (int m = 0; m < Mm; m++) dgv[m] = dGb[(size_t)m * Dd + is];

  float prods[20];
  int p = 0;
  #pragma unroll
  for (int m = 0; m < Mm; m++)
    #pragma unroll
    for (int n = m; n < Mm; n++) prods[p++] = dgv[m] * dgv[n];
  #pragma unroll
  for (int m = 0; m < Mm; m++) prods[p++] = dgv[m] * gv;

  __shared__ float red[20][8];
  __shared__ float tot[20];
  int lane = t & 31, w = t >> 5;
  #pragma unroll
  for (int j = 0; j < 20; j++) {
    float v = prods[j];
    #pragma unroll
    for (int o = 16; o > 0; o >>= 1) v += __shfl_down(v, o, 32);
    if (lane == 0) red[j][w] = v;
  }
  __syncthreads();
  if (t < 20) {
    float s = 0.f;
    #pragma unroll
    for (int w2 = 0; w2 < 8; w2++) s += red[t][w2];
    tot[t] = s;
  }
  __syncthreads();
  if (t == 0) {
    float A[Mm][Mm], rhs[Mm], al[Mm];
    int q = 0;
    for (int m = 0; m < Mm; m++)
      for (int n = m; n < Mm; n++) { float v = tot[q++]; A[m][n] = v; A[n][m] = v; }
    for (int m = 0; m < Mm; m++) { A[m][m] += REGf; rhs[m] = tot[15 + m]; }
    for (int i = 0; i < Mm; i++) {
      float inv = 1.0f / A[i][i];
      for (int j = i; j < Mm; j++) A[i][j] *= inv;
      rhs[i] *= inv;
      for (int r = i + 1; r < Mm; r++) {
        float f = A[r][i];
        for (int j = i; j < Mm; j++) A[r][j] -= f * A[i][j];
        rhs[r] -= f * rhs[i];
      }
    }
    for (int i = Mm - 1; i >= 0; i--) {
      float a = rhs[i];
      for (int j = i + 1; j < Mm; j++) a -= A[i][j] * al[j];
      al[i] = a;
    }
    for (int m = 0; m < Mm; m++) alpha[(size_t)b * Mm + m] = al[m];
  }
}

__global__ void zupd(const int* halt, const float* __restrict__ z,
                     const float* __restrict__ g, const float* __restrict__ dX,
                     const float* __restrict__ dG, const float* __restrict__ alpha,
                     float* __restrict__ znew)
{
  if (*halt) return;
  size_t i = (size_t)blockIdx.x * blockDim.x + threadIdx.x;
  if (i >= (size_t)Bsz * Dd) return;
  size_t b = i / Dd, d = i % Dd;
  const float* al = alpha + b * Mm;
  float sx = 0.f, sg = 0.f;
  #pragma unroll
  for (int m = 0; m < Mm; m++) {
    size_t o = (b * Mm + m) * Dd + d;
    sx += dX[o] * al[m];
    sg += dG[o] * al[m];
  }
  znew[i] = z[i] - sx + (g[i] - sg);
}

__global__ void commit_simple(const int* halt, float* __restrict__ z,
                              float* __restrict__ pz, float* __restrict__ pg,
                              const float* __restrict__ g,
                              const float* __restrict__ znew)
{
  if (*halt) return;
  size_t i = (size_t)blockIdx.x * blockDim.x + threadIdx.x;
  if (i >= (size_t)Bsz * Dd) return;
  pz[i] = z[i]; z[i] = znew[i]; pg[i] = g[i];
}

__global__ void commit_safe(const int* halt, const int* SI, float* __restrict__ z,
                            float* __restrict__ pz, float* __restrict__ pg,
                            const float* __restrict__ g,
                            const float* __restrict__ znew)
{
  if (*halt) return;
  int bad = SI[4];
  size_t i = (size_t)blockIdx.x * blockDim.x + threadIdx.x;
  if (i >= (size_t)Bsz * Dd) return;
  float zacc = bad ? (z[i] + g[i]) : znew[i];
  pz[i] = zacc; z[i] = zacc; pg[i] = g[i];
}

__global__ void bump(int* SI) {
  if (threadIdx.x == 0 && !SI[2]) SI[1]++;
}

__global__ void init_state(const float* __restrict__ x, const float* __restrict__ bvec,
                           float* __restrict__ z, float* __restrict__ pz,
                           float* __restrict__ pg, float* __restrict__ bz)
{
  size_t i = (size_t)blockIdx.x * blockDim.x + threadIdx.x;
  if (i >= (size_t)Bsz * Dd) return;
  int d = (int)(i % Dd);
  float g0 = tanhf(x[i] + bvec[d]);   // f(0) - 0
  z[i] = g0; pz[i] = 0.f; pg[i] = g0; bz[i] = 0.f;
}

__global__ void zero_mem(float* p, size_t n) {
  size_t i = (size_t)blockIdx.x * blockDim.x + threadIdx.x;
  if (i < n) p[i] = 0.f;
}

__global__ void init_scalars(float* S, int* SI, int* sidx) {
  int t = threadIdx.x;
  if (t == 0) {
    S[0] = __builtin_inff(); S[1] = __builtin_inff();
    SI[0] = 0; SI[1] = 0; SI[2] = 0; SI[3] = 0; SI[4] = 0;
  }
  // deterministic sketch indices (hash; jax PRNG not bit-reproducible here)
  unsigned h = (unsigned)t * 747796405u + 2891336453u;
  h ^= h >> 17; h *= 0x9E3779B1u; h ^= h >> 13;
  sidx[t] = (int)(h & (Dd - 1));
}

__global__ void finalize(const float* S, const float* __restrict__ z,
                         const float* __restrict__ bz, float* __restrict__ out)
{
  size_t i = (size_t)blockIdx.x * blockDim.x + threadIdx.x;
  if (i >= (size_t)Bsz * Dd) return;
  out[i] = (S[1] < TOLf) ? z[i] : bz[i];
}

// ---------------------------------------------------------------------------
extern "C" void kernel_launch(void* const* d_in, const int* in_sizes, int n_in,
                              void* d_out, int out_size, void* d_ws, size_t ws_size,
                              hipStream_t stream)
{
  (void)in_sizes; (void)n_in; (void)out_size; (void)ws_size;
  const float* x    = (const float*)d_in[0];
  const float* W    = (const float*)d_in[1];
  const float* bvec = (const float*)d_in[2];
  float* out = (float*)d_out;

  const size_t N = (size_t)Bsz * Dd;          // 2,097,152
  float* ws    = (float*)d_ws;
  float* z     = ws + 0 * N;
  float* pz    = ws + 1 * N;
  float* pg    = ws + 2 * N;
  float* g     = ws + 3 * N;
  float* g2    = ws + 4 * N;
  float* znew  = ws + 5 * N;
  float* bz    = ws + 6 * N;
  float* dX    = ws + 7 * N;                  // B x M x D  (5N)
  float* dG    = ws + 12 * N;                 // B x M x D  (5N)
  float* gn    = ws + 17 * N;
  float* zn    = gn + Bsz;
  float* gn2   = zn + Bsz;
  float* alpha = gn2 + Bsz;                   // B x M
  float* S     = alpha + (size_t)Bsz * Mm;    // 2 floats: bres, rel
  int*   SI    = (int*)(S + 2);               // done, ptr, halt, better, bad
  int*   sidx  = SI + 8;                      // 256 sketch indices

  const int T = 256;
  const int GBD = (int)(N / T);               // 8192 blocks for B*D elementwise
  dim3 gGemm(Dd / 128, Bsz / 64);             // 8 x 32 tiles

  // --- init ---
  init_state<<<GBD, T, 0, stream>>>(x, bvec, z, pz, pg, bz);
  zero_mem<<<(int)((10 * N + T - 1) / T), T, 0, stream>>>(dX, 10 * N); // dX+dG
  init_scalars<<<1, 256, 0, stream>>>(S, SI, sidx);

  // --- fori_loop(1, MAX_ITER) : 11 iterations ---
  for (int k = 1; k < 12; ++k) {
    gemm_g<<<gGemm, T, 0, stream>>>(&SI[0], z, W, x, bvec, g);      // g = f(z)-z
    rownorm<<<Bsz, T, 0, stream>>>(&SI[0], g, z, gn, zn);
    control1<<<1, T, 0, stream>>>(gn, zn, S, SI);                   // bres/rel/halt
    bz_copy<<<GBD, T, 0, stream>>>(&SI[3], z, bz);
    colupd<<<GBD, T, 0, stream>>>(&SI[2], SI, z, pz, g, pg, dX, dG);
    sketch_solve<<<Bsz, T, 0, stream>>>(&SI[2], dG, g, sidx, alpha);
    zupd<<<GBD, T, 0, stream>>>(&SI[2], z, g, dX, dG, alpha, znew);
    if (k > 2) {
      gemm_g<<<gGemm, T, 0, stream>>>(&SI[2], znew, W, x, bvec, g2); // safeguard
      rownorm<<<Bsz, T, 0, stream>>>(&SI[2], g2, (const float*)nullptr, gn2,
                                     (float*)nullptr);
      control2<<<1, T, 0, stream>>>(&SI[2], gn2, S, SI);
      commit_safe<<<GBD, T, 0, stream>>>(&SI[2], SI, z, pz, pg, g, znew);
    } else {
      commit_simple<<<GBD, T, 0, stream>>>(&SI[2], z, pz, pg, g, znew);
    }
    bump<<<1, 32, 0, stream>>>(SI);
  }

  finalize<<<GBD, T, 0, stream>>>(S, z, bz, out);
}